// MambaBlock_2018634629637
// MI455X (gfx1250) — compile-verified
//
#include <hip/hip_runtime.h>
#include <hip/hip_bf16.h>

#define DIM     1024
#define D_INNER 2048
#define D_STATE 16
#define D_CONV  4
#define BATCH   4
#define SEQ     2048
#define NTOK    (BATCH * SEQ)          // 8192 tokens

typedef __attribute__((ext_vector_type(2))) float v2f;
typedef __attribute__((ext_vector_type(8))) float v8f;

// ---------------------------------------------------------------------------
// LayerNorm: one block (256 threads) per token, DIM=1024 -> 4 elems/thread
// ---------------------------------------------------------------------------
__global__ __launch_bounds__(256) void ln_kernel(
    const float* __restrict__ x, const float* __restrict__ gamma,
    const float* __restrict__ beta, float* __restrict__ out)
{
    const int m   = blockIdx.x;
    const int tid = threadIdx.x;
    const float* xr = x + (size_t)m * DIM;

    float v[4], lsum = 0.f, lsq = 0.f;
#pragma unroll
    for (int i = 0; i < 4; ++i) {
        v[i] = xr[tid + 256 * i];
        lsum += v[i];
        lsq  += v[i] * v[i];
    }
    __shared__ float s1[8], s2[8];
    const int wave = tid >> 5, lane = tid & 31;
#pragma unroll
    for (int o = 16; o > 0; o >>= 1) {
        lsum += __shfl_xor(lsum, o);
        lsq  += __shfl_xor(lsq,  o);
    }
    if (lane == 0) { s1[wave] = lsum; s2[wave] = lsq; }
    __syncthreads();
    if (wave == 0) {
        float a = (lane < 8) ? s1[lane] : 0.f;
        float b = (lane < 8) ? s2[lane] : 0.f;
#pragma unroll
        for (int o = 4; o > 0; o >>= 1) { a += __shfl_xor(a, o); b += __shfl_xor(b, o); }
        if (lane == 0) { s1[0] = a; s2[0] = b; }
    }
    __syncthreads();
    const float mean = s1[0] * (1.f / DIM);
    const float var  = s2[0] * (1.f / DIM) - mean * mean;
    const float rstd = rsqrtf(var + 1e-5f);

    float* orow = out + (size_t)m * DIM;
#pragma unroll
    for (int i = 0; i < 4; ++i) {
        const int c = tid + 256 * i;
        orow[c] = (v[i] - mean) * rstd * gamma[c] + beta[c];
    }
}

// ---------------------------------------------------------------------------
// f32 WMMA GEMM: Out[M,N] = A[M,K] * W[N,K]^T (+ optional addend[M,N])
// Block tile 64x128 (8 waves as 2x4), per-wave tile 32x32 (2x2 accumulators
// -> each LDS fragment feeds two v_wmma), double-buffered LDS with register
// staging so HBM loads overlap the matrix pipe. BK=16 -> 16 wmma per chunk.
// ---------------------------------------------------------------------------
#define BM 64
#define BN 128
#define BK 16

__global__ __launch_bounds__(256) void gemm_wmma_f32(
    const float* __restrict__ A, const float* __restrict__ W,
    const float* __restrict__ addend, float* __restrict__ Out,
    int M, int N, int K)
{
    __shared__ float As[2][BM][BK + 1];
    __shared__ float Bs[2][BN][BK + 1];

    const int tid  = threadIdx.x;
    const int wave = tid >> 5;           // 0..7
    const int lane = tid & 31;
    const int wm   = wave >> 2;          // 0..1 (M)
    const int wn   = wave & 3;           // 0..3 (N)
    const int m0   = blockIdx.y * BM;
    const int n0   = blockIdx.x * BN;

    const int laneM = lane & 15;
    const int koff  = (lane >> 4) << 1;  // lanes 0-15: K+0/K+1, lanes 16-31: K+2/K+3
    const int am0   = wm * 32 + laneM;
    const int am1   = am0 + 16;
    const int bn0   = wn * 32 + laneM;
    const int bn1   = bn0 + 16;

    v8f acc00 = {}, acc01 = {}, acc10 = {}, acc11 = {};

    // preload chunk 0 into buffer 0
#pragma unroll
    for (int i = 0; i < (BM * BK) / 256; ++i) {
        const int e = i * 256 + tid;
        As[0][e >> 4][e & 15] = A[(size_t)(m0 + (e >> 4)) * K + (e & 15)];
    }
#pragma unroll
    for (int i = 0; i < (BN * BK) / 256; ++i) {
        const int e = i * 256 + tid;
        Bs[0][e >> 4][e & 15] = W[(size_t)(n0 + (e >> 4)) * K + (e & 15)];
    }
    __syncthreads();

    const int nchunk = K / BK;
    for (int ck = 0; ck < nchunk; ++ck) {
        const int cur = ck & 1, nxt = cur ^ 1;

        // stage next chunk in registers (overlaps with wmma below)
        float ra[(BM * BK) / 256], rb[(BN * BK) / 256];
        if (ck + 1 < nchunk) {
            const int k0 = (ck + 1) * BK;
#pragma unroll
            for (int i = 0; i < (BM * BK) / 256; ++i) {
                const int e = i * 256 + tid;
                ra[i] = A[(size_t)(m0 + (e >> 4)) * K + k0 + (e & 15)];
            }
#pragma unroll
            for (int i = 0; i < (BN * BK) / 256; ++i) {
                const int e = i * 256 + tid;
                rb[i] = W[(size_t)(n0 + (e >> 4)) * K + k0 + (e & 15)];
            }
        }

        // 16 wmma on current buffer; each frag pair feeds two wmma
#pragma unroll
        for (int kk = 0; kk < BK; kk += 4) {
            v2f a0, a1, b0, b1;
            a0[0] = As[cur][am0][kk + koff]; a0[1] = As[cur][am0][kk + koff + 1];
            a1[0] = As[cur][am1][kk + koff]; a1[1] = As[cur][am1][kk + koff + 1];
            b0[0] = Bs[cur][bn0][kk + koff]; b0[1] = Bs[cur][bn0][kk + koff + 1];
            b1[0] = Bs[cur][bn1][kk + koff]; b1[1] = Bs[cur][bn1][kk + koff + 1];
            acc00 = __builtin_amdgcn_wmma_f32_16x16x4_f32(false, a0, false, b0, (short)0, acc00, false, false);
            acc01 = __builtin_amdgcn_wmma_f32_16x16x4_f32(false, a0, false, b1, (short)0, acc01, false, false);
            acc10 = __builtin_amdgcn_wmma_f32_16x16x4_f32(false, a1, false, b0, (short)0, acc10, false, false);
            acc11 = __builtin_amdgcn_wmma_f32_16x16x4_f32(false, a1, false, b1, (short)0, acc11, false, false);
        }

        // commit staged registers to the other buffer
        if (ck + 1 < nchunk) {
#pragma unroll
            for (int i = 0; i < (BM * BK) / 256; ++i) {
                const int e = i * 256 + tid;
                As[nxt][e >> 4][e & 15] = ra[i];
            }
#pragma unroll
            for (int i = 0; i < (BN * BK) / 256; ++i) {
                const int e = i * 256 + tid;
                Bs[nxt][e >> 4][e & 15] = rb[i];
            }
        }
        __syncthreads();
    }

    // C/D layout: VGPR r -> M = r (lanes 0-15) / r+8 (lanes 16-31), N = lane&15
    v8f accs[2][2] = {{acc00, acc01}, {acc10, acc11}};
    const int lm8 = (lane >> 4) << 3;
    if (addend) {
#pragma unroll
        for (int i = 0; i < 2; ++i) {
            const int mb = m0 + wm * 32 + i * 16 + lm8;
#pragma unroll
            for (int j = 0; j < 2; ++j) {
                const int nc = n0 + wn * 32 + j * 16 + laneM;
#pragma unroll
                for (int r = 0; r < 8; ++r) {
                    const size_t idx = (size_t)(mb + r) * N + nc;
                    Out[idx] = accs[i][j][r] + addend[idx];
                }
            }
        }
    } else {
#pragma unroll
        for (int i = 0; i < 2; ++i) {
            const int mb = m0 + wm * 32 + i * 16 + lm8;
#pragma unroll
            for (int j = 0; j < 2; ++j) {
                const int nc = n0 + wn * 32 + j * 16 + laneM;
#pragma unroll
                for (int r = 0; r < 8; ++r)
                    Out[(size_t)(mb + r) * N + nc] = accs[i][j][r];
            }
        }
    }
}

// ---------------------------------------------------------------------------
// Depthwise causal conv (width 4) + SiLU, elementwise over [B,L,D_INNER]
// ---------------------------------------------------------------------------
__global__ __launch_bounds__(256) void conv_silu_kernel(
    const float* __restrict__ h, const float* __restrict__ cw,
    float* __restrict__ out)
{
    const long long idx = (long long)blockIdx.x * 256 + threadIdx.x;
    const int c  = (int)(idx % D_INNER);
    const long long bl = idx / D_INNER;         // b*SEQ + l
    const int l  = (int)(bl % SEQ);

    float acc = 0.f;
#pragma unroll
    for (int k = 0; k < D_CONV; ++k) {
        const int ll = l + k - (D_CONV - 1);
        if (ll >= 0)
            acc += h[(bl + (long long)(k - (D_CONV - 1))) * D_INNER + c] * cw[c * D_CONV + k];
    }
    out[idx] = acc / (1.f + __expf(-acc));       // SiLU
}

// ---------------------------------------------------------------------------
// x_proj: proj[m, 0:33] = h[m, :] @ x_proj_w[n, :]^T, split into dt/B/C
// One block per token; h row staged in LDS; one wave per output column.
// ---------------------------------------------------------------------------
__global__ __launch_bounds__(256) void xproj_kernel(
    const float* __restrict__ h, const float* __restrict__ Wp,
    float* __restrict__ dtin, float* __restrict__ Bp, float* __restrict__ Cp)
{
    __shared__ float sh[D_INNER];
    const int m = blockIdx.x;
    const float* hr = h + (size_t)m * D_INNER;
    for (int i = threadIdx.x; i < D_INNER; i += 256) sh[i] = hr[i];
    __syncthreads();

    const int wave = threadIdx.x >> 5, lane = threadIdx.x & 31;
    for (int n = wave; n < 2 * D_STATE + 1; n += 8) {
        const float* wr = Wp + (size_t)n * D_INNER;
        float s = 0.f;
        for (int k = lane; k < D_INNER; k += 32) s += sh[k] * wr[k];
#pragma unroll
        for (int o = 16; o > 0; o >>= 1) s += __shfl_xor(s, o);
        if (lane == 0) {
            if (n == 0)                dtin[m] = s;
            else if (n <= D_STATE)     Bp[(size_t)m * D_STATE + (n - 1)] = s;
            else                       Cp[(size_t)m * D_STATE + (n - 1 - D_STATE)] = s;
        }
    }
}

// ---------------------------------------------------------------------------
// Selective scan: 32 blocks x 256 threads; thread = (batch, channel).
// 16 states in registers; per-step B/C/dt staged in LDS; SiLU fused on y.
// Next-step activation row prefetched (global_prefetch_b8) into L2/L0 while
// the current step's recurrence executes. Final state -> tail of d_out.
// ---------------------------------------------------------------------------
__global__ __launch_bounds__(256) void scan_kernel(
    const float* __restrict__ xconv,  // [B,L,D_INNER] post conv+silu
    const float* __restrict__ dtin,   // [B*L]
    const float* __restrict__ Bp,     // [B*L, D_STATE]
    const float* __restrict__ Cp,     // [B*L, D_STATE]
    const float* __restrict__ dt_w,   // [D_INNER] (dt_proj_w[:,0])
    const float* __restrict__ dt_b,   // [D_INNER]
    const float* __restrict__ A_log,  // [D_INNER, D_STATE]
    const float* __restrict__ Dvec,   // [D_INNER]
    float* __restrict__ ysilu,        // [B,L,D_INNER]
    float* __restrict__ state_out)    // [B, D_INNER, D_STATE]
{
    const int b = blockIdx.x >> 3;
    const int c = ((blockIdx.x & 7) << 8) + threadIdx.x;

    float negA[D_STATE], st[D_STATE];
#pragma unroll
    for (int s = 0; s < D_STATE; ++s) {
        negA[s] = -__expf(A_log[(size_t)c * D_STATE + s]);
        st[s] = 0.f;
    }
    const float wdt = dt_w[c], bdt = dt_b[c], Dc = Dvec[c];

    __shared__ float sB[D_STATE], sC[D_STATE], sdt;
    const size_t base = (size_t)b * SEQ;

    for (int t = 0; t < SEQ; ++t) {
        const size_t tok = base + t;
        if (threadIdx.x < D_STATE)
            sB[threadIdx.x] = Bp[tok * D_STATE + threadIdx.x];
        else if (threadIdx.x < 2 * D_STATE)
            sC[threadIdx.x - D_STATE] = Cp[tok * D_STATE + (threadIdx.x - D_STATE)];
        if (threadIdx.x == 2 * D_STATE) sdt = dtin[tok];
        __syncthreads();

        const float dtr = sdt * wdt + bdt;
        const float dt  = (dtr > 20.f) ? dtr : log1pf(__expf(dtr));   // softplus
        const float xt  = xconv[tok * D_INNER + c];
        if (t + 1 < SEQ)                                              // hide next step's HBM/L2 latency
            __builtin_prefetch(&xconv[(tok + 1) * D_INNER + c], 0, 3);

        float y = Dc * xt;
#pragma unroll
        for (int s = 0; s < D_STATE; ++s) {
            st[s] = __expf(negA[s] * dt) * st[s] + sB[s] * xt;
            y += st[s] * sC[s];
        }
        ysilu[tok * D_INNER + c] = y / (1.f + __expf(-y));            // SiLU fused
        __syncthreads();
    }
#pragma unroll
    for (int s = 0; s < D_STATE; ++s)
        state_out[((size_t)b * D_INNER + c) * D_STATE + s] = st[s];
}

// ---------------------------------------------------------------------------
extern "C" void kernel_launch(void* const* d_in, const int* in_sizes, int n_in,
                              void* d_out, int out_size, void* d_ws, size_t ws_size,
                              hipStream_t stream)
{
    const float* x         = (const float*)d_in[0];   // [B,SEQ,DIM]
    const float* in_proj_w = (const float*)d_in[1];   // [D_INNER, DIM]
    const float* conv_w    = (const float*)d_in[2];   // [D_INNER, D_CONV]
    const float* x_proj_w  = (const float*)d_in[3];   // [33, D_INNER]
    const float* dt_proj_w = (const float*)d_in[4];   // [D_INNER, 1]
    const float* dt_proj_b = (const float*)d_in[5];   // [D_INNER]
    const float* A_log     = (const float*)d_in[6];   // [D_INNER, D_STATE]
    const float* Dvec      = (const float*)d_in[7];   // [D_INNER]
    const float* out_proj_w= (const float*)d_in[8];   // [DIM, D_INNER]
    const float* norm_w    = (const float*)d_in[9];   // [DIM]
    const float* norm_b    = (const float*)d_in[10];  // [DIM]

    float* out_main  = (float*)d_out;                           // [B,SEQ,DIM]
    float* out_state = out_main + (size_t)NTOK * DIM;           // [B,D_INNER,D_STATE]

    // workspace layout (floats)
    float* ws    = (float*)d_ws;
    float* h_ln  = ws;                                          // [NTOK, DIM]
    float* h1    = h_ln + (size_t)NTOK * DIM;                   // [NTOK, D_INNER] (in_proj out, reused as y_silu)
    float* h2    = h1 + (size_t)NTOK * D_INNER;                 // [NTOK, D_INNER] (conv+silu out)
    float* dtin  = h2 + (size_t)NTOK * D_INNER;                 // [NTOK]
    float* Bpar  = dtin + NTOK;                                 // [NTOK, D_STATE]
    float* Cpar  = Bpar + (size_t)NTOK * D_STATE;               // [NTOK, D_STATE]

    // 1) LayerNorm
    ln_kernel<<<NTOK, 256, 0, stream>>>(x, norm_w, norm_b, h_ln);

    // 2) in_proj GEMM: [8192,1024] x [1024,2048] -> h1
    gemm_wmma_f32<<<dim3(D_INNER / BN, NTOK / BM), 256, 0, stream>>>(
        h_ln, in_proj_w, nullptr, h1, NTOK, D_INNER, DIM);

    // 3) depthwise conv + SiLU -> h2
    conv_silu_kernel<<<(NTOK * D_INNER) / 256, 256, 0, stream>>>(h1, conv_w, h2);

    // 4) x_proj skinny GEMM -> dt/B/C
    xproj_kernel<<<NTOK, 256, 0, stream>>>(h2, x_proj_w, dtin, Bpar, Cpar);

    // 5) selective scan (+fused softplus, SiLU); y_silu overwrites h1
    scan_kernel<<<BATCH * (D_INNER / 256), 256, 0, stream>>>(
        h2, dtin, Bpar, Cpar, dt_proj_w, dt_proj_b, A_log, Dvec, h1, out_state);

    // 6) out_proj GEMM + residual: [8192,2048] x [2048,1024] + x -> d_out
    gemm_wmma_f32<<<dim3(DIM / BN, NTOK / BM), 256, 0, stream>>>(
        h1, out_proj_w, x, out_main, NTOK, DIM, D_INNER);
}